// ACSL_65111704207420
// MI455X (gfx1250) — compile-verified
//
#include <hip/hip_runtime.h>
#include <hip/hip_bf16.h>
#include <stdint.h>

// Problem dims (from reference)
#define N_I   1024
#define N_CLS 201          // NUM_CLASSES + 1
#define T_LEN 100
#define ROW_ELEMS (N_CLS * T_LEN)   // 20100
// sigmoid(x) >= 0.3  <=>  x >= log(0.3/0.7)
#define THR_LOGIT (-0.8472978603872034f)
#define LOG2E 1.4426950408889634f
#define LN2   0.6931471805599453f

typedef float v2f __attribute__((ext_vector_type(2)));
typedef float v8f __attribute__((ext_vector_type(8)));

// ---------------- Threefry2x32 (exact JAX semantics) ----------------
__device__ __forceinline__ void tf_round(uint32_t& x0, uint32_t& x1, int r) {
  x0 += x1;
  x1 = (x1 << r) | (x1 >> (32 - r));
  x1 ^= x0;
}

__device__ __forceinline__ uint2 threefry2x32(uint32_t k0, uint32_t k1,
                                              uint32_t x0, uint32_t x1) {
  uint32_t k2 = k0 ^ k1 ^ 0x1BD11BDAu;
  x0 += k0; x1 += k1;
  tf_round(x0, x1, 13); tf_round(x0, x1, 15); tf_round(x0, x1, 26); tf_round(x0, x1, 6);
  x0 += k1; x1 += k2 + 1u;
  tf_round(x0, x1, 17); tf_round(x0, x1, 29); tf_round(x0, x1, 16); tf_round(x0, x1, 24);
  x0 += k2; x1 += k0 + 2u;
  tf_round(x0, x1, 13); tf_round(x0, x1, 15); tf_round(x0, x1, 26); tf_round(x0, x1, 6);
  x0 += k0; x1 += k1 + 3u;
  tf_round(x0, x1, 17); tf_round(x0, x1, 29); tf_round(x0, x1, 16); tf_round(x0, x1, 24);
  x0 += k1; x1 += k2 + 4u;
  tf_round(x0, x1, 13); tf_round(x0, x1, 15); tf_round(x0, x1, 26); tf_round(x0, x1, 6);
  x0 += k2; x1 += k0 + 5u;
  return make_uint2(x0, x1);
}

__device__ __forceinline__ float u01(uint32_t bits) {
  // JAX uniform: ((bits>>9)|0x3f800000).view(f32) - 1.0
  return __uint_as_float((bits >> 9) | 0x3f800000u) - 1.0f;
}

// softplus via raw hardware transcendentals: arg of log is in [1,2], so no
// denorm/inf fixups are needed (v_exp_f32 / v_log_f32 are exact enough here).
__device__ __forceinline__ float softplus_f(float x) {
  const float e = __builtin_amdgcn_exp2f(-fabsf(x) * LOG2E);  // e^{-|x|}
  const float l = __builtin_amdgcn_logf(1.0f + e) * LN2;      // log1p(e)
  return fmaxf(x, 0.0f) + l;
}

// ---------------- Kernel 1: RNG (u1,u2 for snippet t=99) ----------------
__global__ void ACSL_rng_kernel(float* __restrict__ u1, float* __restrict__ u2,
                                float* __restrict__ out) {
  int i = threadIdx.x;  // 512 threads
  if (i == 0) out[0] = 0.0f;  // zero the scalar output each call
  // key(42) = (0,42). split(key,100): counts=iota(200), pairs (j, j+100),
  // res[j] = out0(j) for j<100 else out1(j-100); keys[t]=(res[2t],res[2t+1]).
  // keys[99] = (res[198], res[199]) = (tf(98,198).y, tf(99,199).y)
  uint2 e1 = threefry2x32(0u, 42u, 98u, 198u);
  uint2 e2 = threefry2x32(0u, 42u, 99u, 199u);
  uint32_t K0 = e1.y, K1 = e2.y;  // key[99]
  // split(key99, 2): counts=iota(4), pairs (0,2),(1,3):
  // k1 = (tf(0,2).x, tf(1,3).x) ; k2 = (tf(0,2).y, tf(1,3).y)
  uint2 f1 = threefry2x32(K0, K1, 0u, 2u);
  uint2 f2 = threefry2x32(K0, K1, 1u, 3u);
  // uniform(k, (1024,)): counts=iota(1024), pairs (i, 512+i);
  // bits[i]=out.x, bits[512+i]=out.y
  uint2 g = threefry2x32(f1.x, f2.x, (uint32_t)i, (uint32_t)(i + 512));
  u1[i]       = u01(g.x);
  u1[i + 512] = u01(g.y);
  uint2 h = threefry2x32(f1.y, f2.y, (uint32_t)i, (uint32_t)(i + 512));
  u2[i]       = u01(h.x);
  u2[i + 512] = u01(h.y);
}

// ---------------- Kernel 2: bg subsample selection at t=99 ----------------
__global__ __launch_bounds__(1024) void ACSL_select_kernel(
    const float* __restrict__ labels, const float* __restrict__ u1,
    const float* __restrict__ u2, int* __restrict__ selr, int* __restrict__ selc) {
  __shared__ float u1s[N_I];
  __shared__ float u2s[N_I];
  __shared__ int nbg_s;
  int i = threadIdx.x;  // 1024 threads, one per row
  // argmax over classes at t=99 (first-max wins, like jnp.argmax)
  const float* p = labels + (size_t)i * ROW_ELEMS + (T_LEN - 1);
  float best = p[0];
  int bi = 0;
  for (int c = 1; c < N_CLS; ++c) {
    float v = p[(size_t)c * T_LEN];
    if (v > best) { best = v; bi = c; }
  }
  bool isbg = (bi == N_CLS - 1);
  if (i == 0) nbg_s = 0;
  __syncthreads();
  if (isbg) atomicAdd(&nbg_s, 1);
  u1s[i] = isbg ? u1[i] : __builtin_inff();
  u2s[i] = isbg ? u2[i] : __builtin_inff();
  __syncthreads();
  int m1 = nbg_s / 100;  // int(n_bg * 0.01)
  int m2 = nbg_s / 10;   // int(n_bg * 0.10)
  float a = u1s[i], b = u2s[i];
  int r1 = 0, r2 = 0;
  for (int j = 0; j < N_I; ++j) {  // stable-argsort rank: count (<) with index tie-break
    float x = u1s[j]; r1 += ((x < a) || (x == a && j < i)) ? 1 : 0;
    float y = u2s[j]; r2 += ((y < b) || (y == b && j < i)) ? 1 : 0;
  }
  selr[i] = (isbg && r1 < m1) ? 1 : 0;
  selc[i] = (isbg && r2 < m2) ? 1 : 0;
}

// ---------------- Kernel 3: main streaming loss ----------------
// One block per row i. Phase 1: lab[t] = argmax_c labels[i,c,t].
// Phase 2: per 16-class block per wave, WMMA f32 16x16x4 with all-ones B folds
// sum_t (softplus(x) - [lab[t]==c]*x). Branch-free inner loop: invalid class
// lanes use a clamped row pointer and a 0/1 mask, so EXEC stays all-ones with
// no exec save/restore around the loads, as WMMA requires.
__global__ __launch_bounds__(256) void ACSL_loss_kernel(
    const float* __restrict__ logits, const float* __restrict__ labels,
    const int* __restrict__ selr, const int* __restrict__ selc,
    float* __restrict__ out) {
  __shared__ alignas(16) int lab_s[T_LEN];
  __shared__ float wsum[8];
  const int i = blockIdx.x;
  const int tid = threadIdx.x;
  const size_t rowbase = (size_t)i * ROW_ELEMS;

  // ---- Phase 1: per-t argmax over classes (lanes = t, coalesced over t) ----
  if (tid < T_LEN) {
    const float* p = labels + rowbase + tid;
    float best = p[0];
    int bi = 0;
#pragma unroll 4
    for (int c = 1; c < N_CLS; ++c) {
      float v = p[(size_t)c * T_LEN];
      if (v > best) { best = v; bi = c; }
    }
    lab_s[tid] = bi;
  }
  __syncthreads();

  const int lab99 = lab_s[T_LEN - 1];
  const bool bg = (lab99 == N_CLS - 1);
  const int sr = selr[i];
  const int sc = selc[i];

  const int wave = tid >> 5;
  const int lane = tid & 31;
  const int mrow = lane & 15;   // A-matrix row (class within 16-block)
  const int khalf = lane >> 4;  // K-half owned by this lane

  const v2f Bones = {1.0f, 1.0f};  // all-ones B => D[m,n] = sum_k A[m,k]
  float partial = 0.0f;

  // 13 blocks of 16 classes cover 201 (+7 masked zero rows)
  for (int cb = wave; cb < 13; cb += 8) {
    const int c = cb * 16 + mrow;
    const float msk = (c < N_CLS) ? 1.0f : 0.0f;
    const int cl = (c < N_CLS) ? c : (N_CLS - 1);     // clamped, always valid
    const float* xr = logits + rowbase + (size_t)cl * T_LEN;

    // streaming hint for this wave's next class block (+128 rows), clamped
    {
      const int cn = (cl + 128 < N_CLS) ? (cl + 128) : cl;
      __builtin_prefetch(logits + rowbase + (size_t)cn * T_LEN + 4 * khalf, 0, 3);
    }

    v8f acc = {};
    // 12 iterations x 8 t's: one b128 load + one int4 LDS load -> 2 WMMAs.
    // khalf=0 lanes carry t0..t0+3 (K=0,1 over two wmmas),
    // khalf=1 lanes carry t0+4..t0+7 (K=2,3). Union per pair = t0..t0+7.
    for (int t0 = 0; t0 < 96; t0 += 8) {
      const int ta = t0 + 4 * khalf;
      const float4 xv = *(const float4*)(xr + ta);
      const int4 lv = *(const int4*)(lab_s + ta);
      const float a0 = msk * (softplus_f(xv.x) - ((lv.x == c) ? xv.x : 0.0f));
      const float a1 = msk * (softplus_f(xv.y) - ((lv.y == c) ? xv.y : 0.0f));
      const float a2 = msk * (softplus_f(xv.z) - ((lv.z == c) ? xv.z : 0.0f));
      const float a3 = msk * (softplus_f(xv.w) - ((lv.w == c) ? xv.w : 0.0f));
      const v2f A0 = {a0, a1};
      const v2f A1 = {a2, a3};
      acc = __builtin_amdgcn_wmma_f32_16x16x4_f32(
          false, A0, false, Bones, (short)0, acc, false, false);
      acc = __builtin_amdgcn_wmma_f32_16x16x4_f32(
          false, A1, false, Bones, (short)0, acc, false, false);
    }
    // tail: t = 96..99 (khalf=0 -> 96,97 ; khalf=1 -> 98,99)
    {
      const int ta = 96 + 2 * khalf;
      const float2 xv = *(const float2*)(xr + ta);
      const int la = lab_s[ta];
      const int lb = lab_s[ta + 1];
      const float a0 = msk * (softplus_f(xv.x) - ((la == c) ? xv.x : 0.0f));
      const float a1 = msk * (softplus_f(xv.y) - ((lb == c) ? xv.y : 0.0f));
      const v2f A = {a0, a1};
      acc = __builtin_amdgcn_wmma_f32_16x16x4_f32(
          false, A, false, Bones, (short)0, acc, false, false);
    }

    // D layout: lanes 0-15 hold rows M=0..7 in acc[0..7]; lanes 16-31 rows 8..15.
    // Columns are identical -> lanes 0 and 16 fold the 16 class sums.
    if (mrow == 0) {
      const int cbase = cb * 16 + 8 * khalf;
      for (int v = 0; v < 8; ++v) {
        const int cc = cbase + v;
        if (cc < N_CLS) {
          float w;
          if (!bg) {
            const float x99 = logits[rowbase + (size_t)cc * T_LEN + (T_LEN - 1)];
            w = (cc == lab99 || x99 >= THR_LOGIT) ? 1.0f : 0.0f;
          } else {
            // FREQ classes [150,200) + BG col 200 always; rare/common if selected
            w = ((cc >= 150) || (sr && cc < 50) || (sc && cc >= 50 && cc < 150))
                    ? 1.0f : 0.0f;
          }
          partial += w * acc[v];
        }
      }
    }
  }

  // reduce: only lanes 0 and 16 hold nonzero partials
  partial += __shfl_down(partial, 16, 32);
  if (lane == 0) wsum[wave] = partial;
  __syncthreads();
  if (tid == 0) {
    float tot = 0.0f;
    for (int w = 0; w < 8; ++w) tot += wsum[w];
    atomicAdd(out, tot * (1.0f / (float)(N_I * T_LEN)));
  }
}

extern "C" void kernel_launch(void* const* d_in, const int* in_sizes, int n_in,
                              void* d_out, int out_size, void* d_ws, size_t ws_size,
                              hipStream_t stream) {
  (void)in_sizes; (void)n_in; (void)out_size; (void)ws_size;
  const float* logits = (const float*)d_in[0];  // cls_logits_ [1024,201,100]
  const float* labels = (const float*)d_in[1];  // labels_     [1024,201,100]
  float* out = (float*)d_out;                   // scalar f32

  float* u1 = (float*)d_ws;        // [1024]
  float* u2 = u1 + N_I;            // [1024]
  int* selr = (int*)(u2 + N_I);    // [1024]
  int* selc = selr + N_I;          // [1024]

  hipLaunchKernelGGL(ACSL_rng_kernel, dim3(1), dim3(512), 0, stream, u1, u2, out);
  hipLaunchKernelGGL(ACSL_select_kernel, dim3(1), dim3(N_I), 0, stream,
                     labels, u1, u2, selr, selc);
  hipLaunchKernelGGL(ACSL_loss_kernel, dim3(N_I), dim3(256), 0, stream,
                     logits, labels, selr, selc, out);
}